// VeroneseDecoding_45569603011216
// MI455X (gfx1250) — compile-verified
//
#include <hip/hip_runtime.h>
#include <hip/hip_bf16.h>

// ---------------------------------------------------------------------------
// Fused MLP:  out[:,4:132] = relu(z @ W1 + b1) @ W2 + b2
// Quat head:  out[:,0:4]   = sign-fixed dominant eigenvector of 4x4 sym(z[:,:10])
//
// B=65536, D_IN=256, D_HID=1024, D_OUT=128, OUT_F=132, all float32.
// WMMA: V_WMMA_F32_16X16X4_F32 (full f32 precision, matches reference dtype).
// W1/W2 chunks are stored TRANSPOSED in LDS so each B-fragment is a single
// contiguous 8B ds_load_b64 into an even VGPR pair (no v_mov reassembly),
// with row strides chosen (stride mod 64 == 4) for conflict-free reads.
// ---------------------------------------------------------------------------

typedef __attribute__((ext_vector_type(2))) float v2f;
typedef __attribute__((ext_vector_type(8))) float v8f;

#define B_TOTAL   65536
#define D_IN      256
#define D_HID     1024
#define D_OUT     128
#define OUT_F     132
#define BM        128      // rows per workgroup (8 waves x 16 rows)
#define K2C       64       // hidden-dim chunk per iteration (16 chunks total)
#define NCHUNK    (D_HID / K2C)

// LDS layout (floats). Transposed chunks: wt[col][k].
#define W1T_LD    260      // w1t: [64 cols][260]  (256 k + 4 pad; 260%64==4)
#define W2T_LD    68       // w2t: [128 cols][68]  (64 k + 4 pad;  68%64==4)
#define HST_LD    68       // per-wave h staging [16][68]
#define SMEM_W1T  (K2C * W1T_LD)                  // 16640
#define SMEM_W2T  (D_OUT * W2T_LD)                //  8704
#define SMEM_HST  (8 * 16 * HST_LD)               //  8704
#define SMEM_FLOATS (SMEM_W1T + SMEM_W2T + SMEM_HST)
#define SMEM_BYTES  (SMEM_FLOATS * 4)             // 136192 B < 320KB/WGP

__global__ __launch_bounds__(256) void fused_mlp_kernel(
    const float* __restrict__ z,   // [B][256]
    const float* __restrict__ W1,  // [256][1024]
    const float* __restrict__ b1,  // [1024]
    const float* __restrict__ W2,  // [1024][128]
    const float* __restrict__ b2,  // [128]
    float* __restrict__ out)       // [B][132]
{
    extern __shared__ float smem[];
    float* w1t = smem;                       // W1 chunk^T, [64][W1T_LD]
    float* w2t = smem + SMEM_W1T;            // W2 chunk^T, [128][W2T_LD]
    float* hst = smem + SMEM_W1T + SMEM_W2T; // relu(h) staging, per wave [16][HST_LD]

    const int t    = threadIdx.x;
    const int wave = t >> 5;
    const int lane = t & 31;
    const int half = lane >> 4;   // 0: lanes 0-15, 1: lanes 16-31
    const int ln   = lane & 15;   // N (and M for A-frags) index within fragment

    const int rowbase = blockIdx.x * BM + wave * 16;
    float* hw = hst + wave * (16 * HST_LD);

    // Output accumulators: 16 rows x 128 cols = 8 WMMA C tiles (64 VGPRs).
    v8f acc[8];
    #pragma unroll
    for (int i = 0; i < 8; ++i) {
        v8f zero = {0.f, 0.f, 0.f, 0.f, 0.f, 0.f, 0.f, 0.f};
        acc[i] = zero;
    }

    // z A-fragment base for this lane: row (rowbase + ln), contiguous K pairs.
    const float* zrow = z + (size_t)(rowbase + ln) * D_IN;

    for (int c = 0; c < NCHUNK; ++c) {
        const int k2base = c * K2C;

        // --- cooperative load: W1[:, k2base:+64] -> w1t (transposed) -------
        // 256x64 floats = 4096 float4 over 256 threads = 16 each; coalesced
        // row-major global reads, 4 scalar transposed LDS stores each.
        #pragma unroll
        for (int i = 0; i < 16; ++i) {
            const int idx = i * 256 + t;
            const int r   = idx >> 4;          // k index (W1 row), 0..255
            const int c4  = (idx & 15) << 2;   // chunk-local col, 0..60
            float4 v = *(const float4*)(W1 + (size_t)r * D_HID + k2base + c4);
            w1t[(c4 + 0) * W1T_LD + r] = v.x;
            w1t[(c4 + 1) * W1T_LD + r] = v.y;
            w1t[(c4 + 2) * W1T_LD + r] = v.z;
            w1t[(c4 + 3) * W1T_LD + r] = v.w;
        }
        // --- cooperative load: W2[k2base:+64, :] -> w2t (transposed) -------
        // 64x128 floats = 2048 float4 over 256 threads = 8 each
        #pragma unroll
        for (int i = 0; i < 8; ++i) {
            const int idx = i * 256 + t;
            const int r   = idx >> 5;          // chunk-local k, 0..63
            const int c4  = (idx & 31) << 2;   // output col, 0..124
            float4 v = *(const float4*)(W2 + (size_t)(k2base + r) * D_OUT + c4);
            w2t[(c4 + 0) * W2T_LD + r] = v.x;
            w2t[(c4 + 1) * W2T_LD + r] = v.y;
            w2t[(c4 + 2) * W2T_LD + r] = v.z;
            w2t[(c4 + 3) * W2T_LD + r] = v.w;
        }
        __syncthreads();

        // Prefetch next chunk of W1/W2 into cache while we compute this one.
        if (c + 1 < NCHUNK) {
            const int nk2 = (c + 1) * K2C;
            __builtin_prefetch(W1 + (size_t)t * D_HID + nk2, 0, 1);
            __builtin_prefetch(W2 + (size_t)(nk2 + (t >> 2)) * D_OUT + ((t & 3) << 5), 0, 1);
        }

        // --- GEMM1: h_tile = relu(z_tile @ W1_chunk + b1_chunk) ------------
        // 16 rows x 64 cols per wave, 4 live WMMA C tiles; one z A-fragment
        // per k-step reused across all 4 tiles; B-frags are contiguous b64.
        v8f hAcc[4];
        #pragma unroll
        for (int n2 = 0; n2 < 4; ++n2) {
            const float bv = b1[k2base + n2 * 16 + ln];
            v8f bsplat = {bv, bv, bv, bv, bv, bv, bv, bv};
            hAcc[n2] = bsplat;
        }
        #pragma unroll 8
        for (int k1 = 0; k1 < D_IN; k1 += 4) {
            const int kk = k1 + 2 * half;
            // A 16x4 f32: VGPR0 = K{0,2}, VGPR1 = K{1,3}; lane = M
            v2f a = *(const v2f*)(zrow + kk);
            #pragma unroll
            for (int n2 = 0; n2 < 4; ++n2) {
                // B 4x16 f32 (mirrored A layout): lane = N, VGPR pair = K
                v2f b = *(const v2f*)(w1t + (n2 * 16 + ln) * W1T_LD + kk);
                hAcc[n2] = __builtin_amdgcn_wmma_f32_16x16x4_f32(
                    false, a, false, b, (short)0, hAcc[n2], false, false);
            }
        }
        // relu + stage into per-wave LDS in plain (row, col) order
        #pragma unroll
        for (int n2 = 0; n2 < 4; ++n2) {
            const int ncol = n2 * 16 + ln;
            #pragma unroll
            for (int j = 0; j < 8; ++j) {
                float vv = hAcc[n2][j];
                vv = vv > 0.f ? vv : 0.f;
                hw[(j + 8 * half) * HST_LD + ncol] = vv;  // C layout: M=j+8*half
            }
        }
        // hst is private per wave: same-wave DS ordering suffices, no barrier.

        // --- GEMM2: acc += h_tile @ W2_chunk -------------------------------
        #pragma unroll 4
        for (int k2 = 0; k2 < K2C; k2 += 4) {
            const int kk = k2 + 2 * half;
            v2f a = *(const v2f*)(hw + ln * HST_LD + kk);   // A-frag of relu(h)
            #pragma unroll
            for (int ct = 0; ct < 8; ++ct) {
                v2f b = *(const v2f*)(w2t + (ct * 16 + ln) * W2T_LD + kk);
                acc[ct] = __builtin_amdgcn_wmma_f32_16x16x4_f32(
                    false, a, false, b, (short)0, acc[ct], false, false);
            }
        }
        __syncthreads();   // protect w1t/w2t before next chunk's loads
    }

    // --- epilogue: + b2, scatter into out[:, 4:132] ------------------------
    #pragma unroll
    for (int ct = 0; ct < 8; ++ct) {
        const float b2v = b2[ct * 16 + ln];
        #pragma unroll
        for (int j = 0; j < 8; ++j) {
            const int m = j + 8 * half;
            out[(size_t)(rowbase + m) * OUT_F + 4 + ct * 16 + ln] = acc[ct][j] + b2v;
        }
    }
}

// ---------------------------------------------------------------------------
// Quaternion head: dominant eigenvector of symmetric 4x4 built from z[:, :10],
// via Gershgorin-shifted power iteration (targets largest ALGEBRAIC eigenvalue,
// matching eigh's vecs[:, :, -1]). One thread per batch row.
// ---------------------------------------------------------------------------
__global__ __launch_bounds__(256) void quat_kernel(
    const float* __restrict__ z, float* __restrict__ out)
{
    const int row = blockIdx.x * blockDim.x + threadIdx.x;
    const float* zr = z + (size_t)row * D_IN;

    // triu order: (0,0)(0,1)(0,2)(0,3)(1,1)(1,2)(1,3)(2,2)(2,3)(3,3)
    const float a = zr[0], b = zr[1], cc = zr[2], d = zr[3];
    const float e = zr[4], f = zr[5], g = zr[6];
    const float h = zr[7], i = zr[8];
    const float j = zr[9];

    // Gershgorin shift so all eigenvalues become positive, order preserved.
    const float r0 = fabsf(a) + fabsf(b) + fabsf(cc) + fabsf(d);
    const float r1 = fabsf(b) + fabsf(e) + fabsf(f) + fabsf(g);
    const float r2 = fabsf(cc) + fabsf(f) + fabsf(h) + fabsf(i);
    const float r3 = fabsf(d) + fabsf(g) + fabsf(i) + fabsf(j);
    const float sig = fmaxf(fmaxf(r0, r1), fmaxf(r2, r3)) + 1e-6f;
    const float m00 = a + sig, m11 = e + sig, m22 = h + sig, m33 = j + sig;

    float v0 = 0.8147f, v1 = 0.9058f, v2 = 0.1270f, v3 = 0.9134f;
    #pragma unroll 4
    for (int it = 0; it < 64; ++it) {
        const float w0 = m00 * v0 + b   * v1 + cc  * v2 + d   * v3;
        const float w1 = b   * v0 + m11 * v1 + f   * v2 + g   * v3;
        const float w2 = cc  * v0 + f   * v1 + m22 * v2 + i   * v3;
        const float w3 = d   * v0 + g   * v1 + i   * v2 + m33 * v3;
        const float inv = rsqrtf(w0 * w0 + w1 * w1 + w2 * w2 + w3 * w3 + 1e-37f);
        v0 = w0 * inv; v1 = w1 * inv; v2 = w2 * inv; v3 = w3 * inv;
    }
    const float s = (v0 + 1e-9f) >= 0.f ? 1.f : -1.f;
    float* o = out + (size_t)row * OUT_F;
    o[0] = v0 * s; o[1] = v1 * s; o[2] = v2 * s; o[3] = v3 * s;
}

extern "C" void kernel_launch(void* const* d_in, const int* in_sizes, int n_in,
                              void* d_out, int out_size, void* d_ws, size_t ws_size,
                              hipStream_t stream) {
    const float* z  = (const float*)d_in[0];
    const float* W1 = (const float*)d_in[1];
    const float* b1 = (const float*)d_in[2];
    const float* W2 = (const float*)d_in[3];
    const float* b2 = (const float*)d_in[4];
    float* out = (float*)d_out;

    fused_mlp_kernel<<<B_TOTAL / BM, 256, SMEM_BYTES, stream>>>(z, W1, b1, W2, b2, out);
    quat_kernel<<<B_TOTAL / 256, 256, 0, stream>>>(z, out);
}